// CustomLossMinMax_3642132267438
// MI455X (gfx1250) — compile-verified
//
#include <hip/hip_runtime.h>
#include <hip/hip_bf16.h>

// ---------------------------------------------------------------------------
// CustomLossMinMax for MI455X (gfx1250, wave32).
//
// Roofline: reads pos(128MiB incl p) + neg(64MiB) = 192MiB @ 23.3TB/s ~ 8.3us.
// Compute (N*L fma + M*L max + N*M hinge pairs ~ 0.2GFLOP) is negligible, so
// kernels 1/2 are pure streaming reductions with b128 non-temporal loads.
// The pairwise hinge reduction uses V_WMMA_F32_16X16X4_F32 with a ones
// B-matrix as the row-sum engine (mathematically exact, and free since the
// pair phase is only ~1M WMMAs).
// ---------------------------------------------------------------------------

typedef __attribute__((ext_vector_type(2))) float v2f;
typedef __attribute__((ext_vector_type(4))) float v4f;
typedef __attribute__((ext_vector_type(8))) float v8f;

#define WAVE 32

// ------------------------- Kernel 1: wpos[i] = dot(pos[i,:], p[i,:]) -------
// One wave per row; lanes stride over L/4 float4s (fully coalesced b128).
__global__ void wpos_kernel(const float* __restrict__ pos,
                            const float* __restrict__ p,
                            float* __restrict__ wpos, int L) {
  const int wavesPerBlock = blockDim.x / WAVE;
  const int row  = blockIdx.x * wavesPerBlock + (threadIdx.x / WAVE);
  const int lane = threadIdx.x & (WAVE - 1);
  const v4f* pos4 = (const v4f*)(pos + (size_t)row * L);
  const v4f* p4   = (const v4f*)(p   + (size_t)row * L);
  const int n4 = L >> 2;
  v4f acc = {0.f, 0.f, 0.f, 0.f};
  #pragma unroll 4
  for (int idx = lane; idx < n4; idx += WAVE) {
    v4f a = __builtin_nontemporal_load(pos4 + idx);  // read-once stream -> NT
    v4f b = __builtin_nontemporal_load(p4   + idx);
    acc += a * b;                                    // contracts to v_fma
  }
  float s = (acc.x + acc.y) + (acc.z + acc.w);
  #pragma unroll
  for (int off = 16; off >= 1; off >>= 1) s += __shfl_xor(s, off, WAVE);
  if (lane == 0) wpos[row] = s;
}

// ------------------------- Kernel 2: neg_max[j] = max_l neg[j,l] -----------
__global__ void negmax_kernel(const float* __restrict__ neg,
                              float* __restrict__ negmax, int L) {
  const int wavesPerBlock = blockDim.x / WAVE;
  const int row  = blockIdx.x * wavesPerBlock + (threadIdx.x / WAVE);
  const int lane = threadIdx.x & (WAVE - 1);
  const v4f* neg4 = (const v4f*)(neg + (size_t)row * L);
  const int n4 = L >> 2;
  const float ninf = -__builtin_inff();
  v4f acc = {ninf, ninf, ninf, ninf};
  #pragma unroll 4
  for (int idx = lane; idx < n4; idx += WAVE) {
    v4f a = __builtin_nontemporal_load(neg4 + idx);
    acc = __builtin_elementwise_max(acc, a);
  }
  float s = __builtin_fmaxf(__builtin_fmaxf(acc.x, acc.y),
                            __builtin_fmaxf(acc.z, acc.w));
  #pragma unroll
  for (int off = 16; off >= 1; off >>= 1)
    s = __builtin_fmaxf(s, __shfl_xor(s, off, WAVE));
  if (lane == 0) negmax[row] = s;
}

// ------------------------- Kernel 3: pairwise hinge via WMMA ---------------
// Each wave owns 16 rows i. Per step it builds the 16x4 f32 A-tile
//   H[m,k] = relu((1 - wpos[i0+m]) + neg_max[j+k])
// in the exact ISA A-layout (VGPR0: K0 | K2 at lane>=16, VGPR1: K1 | K3) and
// accumulates D = H x ones(4x16) + C, i.e. exact row-sums over j.
template <int JCHUNK>
__global__ void pair_kernel(const float* __restrict__ wpos,
                            const float* __restrict__ negmax,
                            float* __restrict__ partials) {
  __shared__ float s_nm[JCHUNK];
  __shared__ float s_part[8];
  const int tid = threadIdx.x;
  const int j0  = blockIdx.y * JCHUNK;
  for (int t = tid; t < JCHUNK; t += blockDim.x) s_nm[t] = negmax[j0 + t];
  __syncthreads();

  const int w    = tid / WAVE;
  const int lane = tid & (WAVE - 1);
  const int i0   = (blockIdx.x * (blockDim.x / WAVE) + w) * 16;
  const int m    = lane & 15;
  const float cterm = 1.0f - wpos[i0 + m];
  const int kbase = (lane >> 4) << 1;  // lanes 0-15 -> K{0,1}; 16-31 -> K{2,3}

  v8f acc = {0.f, 0.f, 0.f, 0.f, 0.f, 0.f, 0.f, 0.f};
  const v2f ones = {1.0f, 1.0f};
  #pragma unroll 4
  for (int j = 0; j < JCHUNK; j += 4) {
    v2f a;
    a.x = __builtin_fmaxf(cterm + s_nm[j + kbase],     0.0f);
    a.y = __builtin_fmaxf(cterm + s_nm[j + kbase + 1], 0.0f);
    acc = __builtin_amdgcn_wmma_f32_16x16x4_f32(
        /*neg_a=*/false, a, /*neg_b=*/false, ones,
        /*c_mod=*/(short)0, acc, /*reuse_a=*/false, /*reuse_b=*/false);
  }
  // C layout: lanes 0-15 hold rows 0-7 (VGPR r = row r), lanes 16-31 rows
  // 8-15; every column n is identical. Column n=0 lives in lanes 0 and 16.
  float s = ((acc[0] + acc[1]) + (acc[2] + acc[3])) +
            ((acc[4] + acc[5]) + (acc[6] + acc[7]));
  float tot = __shfl(s, 0, WAVE) + __shfl(s, 16, WAVE);
  if (lane == 0) s_part[w] = tot;
  __syncthreads();
  if (tid == 0) {
    float b = 0.f;
    const int nw = blockDim.x / WAVE;
    for (int q = 0; q < nw; ++q) b += s_part[q];
    partials[blockIdx.y * gridDim.x + blockIdx.x] = b;
  }
}

// ------------------------- Kernel 4: final reduce + scale ------------------
__global__ void finish_kernel(const float* __restrict__ partials, int n,
                              float* __restrict__ out, float scale) {
  __shared__ float sh[256];
  float s = 0.f;
  for (int i = threadIdx.x; i < n; i += blockDim.x) s += partials[i];
  sh[threadIdx.x] = s;
  __syncthreads();
  for (int off = blockDim.x >> 1; off > 0; off >>= 1) {
    if ((int)threadIdx.x < off) sh[threadIdx.x] += sh[threadIdx.x + off];
    __syncthreads();
  }
  if (threadIdx.x == 0) out[0] = sh[0] * scale;  // always written (poison-safe)
}

extern "C" void kernel_launch(void* const* d_in, const int* in_sizes, int n_in,
                              void* d_out, int out_size, void* d_ws, size_t ws_size,
                              hipStream_t stream) {
  const int L = 2048;
  const int N = in_sizes[0] / L;   // 8192
  const int M = in_sizes[1] / L;   // 8192

  const float* pos = (const float*)d_in[0];
  const float* neg = (const float*)d_in[1];
  const float* p   = (const float*)d_in[2];
  float* out = (float*)d_out;

  float* wpos     = (float*)d_ws;          // N floats
  float* negmax   = wpos + N;              // M floats
  float* partials = negmax + M;            // gx*gy floats

  const int BLK = 256;                     // 8 waves (wave32)
  const int wavesPerBlock = BLK / WAVE;

  wpos_kernel<<<N / wavesPerBlock, BLK, 0, stream>>>(pos, p, wpos, L);
  negmax_kernel<<<M / wavesPerBlock, BLK, 0, stream>>>(neg, negmax, L);

  constexpr int JCHUNK = 2048;             // 8 KB of LDS for the neg_max tile
  const int gx = N / (wavesPerBlock * 16); // 64
  const int gy = M / JCHUNK;               // 4
  dim3 grid(gx, gy);
  pair_kernel<JCHUNK><<<grid, BLK, 0, stream>>>(wpos, negmax, partials);

  const float scale = 1.0f / ((float)N * (float)M);
  finish_kernel<<<1, 256, 0, stream>>>(partials, gx * gy, out, scale);
}